// Attention_90452011253813
// MI455X (gfx1250) — compile-verified
//
#include <hip/hip_runtime.h>

typedef float v2f __attribute__((ext_vector_type(2)));
typedef float v4f __attribute__((ext_vector_type(4)));
typedef float v8f __attribute__((ext_vector_type(8)));

#define N_TOK  4096
#define C_IN   64
#define HEADS  4
#define DHEAD  4
#define HIDDEN 16
#define BATCH  4

// Workspace layout (floats):
//  QT : [b][h][n][d]  (q, pre-scaled by 0.5*log2e)  size 262144
//  KT : [b][h][n][d]                                size 262144
//  V  : [b][h][d][n]                                size 262144
//  ATT: [b][n][hd]                                  size 262144
#define REGION (BATCH * HEADS * N_TOK * DHEAD)   // 262144

// cross-half (lane L <-> L^16) exchange as a single VALU op:
// v_permlanex16_b32 with identity nibble selects swaps the two 16-lane halves.
static __device__ __forceinline__ float swap16(float x) {
    int i = __float_as_int(x);
    int r = __builtin_amdgcn_permlanex16(i, i, 0x76543210, 0xfedcba98, false, false);
    return __int_as_float(r);
}
static __device__ __forceinline__ float fexp2(float x) {
    return __builtin_amdgcn_exp2f(x);
}

// ---------------- Kernel 1: qkv projection, writes Q/K transposed, V row-major ----------------
__global__ void __launch_bounds__(256) qkv_kernel(const float* __restrict__ x,
                                                  const float* __restrict__ w,
                                                  float* __restrict__ ws) {
    int tid = blockIdx.x * blockDim.x + threadIdx.x;     // = b*48*4096 + o*4096 + n
    int n = tid & (N_TOK - 1);
    int o = (tid >> 12) % 48;
    int b = tid / (N_TOK * 48);
    const float* xr = x + (size_t)b * C_IN * N_TOK + n;
    const float* wr = w + o * C_IN;
    float acc = 0.f;
#pragma unroll 8
    for (int c = 0; c < C_IN; ++c)
        acc = fmaf(wr[c], xr[(size_t)c * N_TOK], acc);

    int oo = o & 15;
    int h = oo >> 2, d = oo & 3;
    size_t bh = (size_t)(b * HEADS + h);
    if (o < 16) {            // Q: fold in scale * log2(e) so scores are base-2 logits
        ws[(bh * N_TOK + n) * DHEAD + d] = acc * 0.72134752044448170368f;
    } else if (o < 32) {     // K: [bh][n][d]
        ws[REGION + (bh * N_TOK + n) * DHEAD + d] = acc;
    } else {                 // V: [bh][d][n]
        ws[2 * REGION + (bh * DHEAD + d) * N_TOK + n] = acc;
    }
}

// ---------------- Kernel 2: flash attention via V_WMMA_F32_16X16X4_F32 ----------------
// One wave owns one 16-query tile; 32 j's per iteration (two S^T tiles).
// S^T(jxi) = K(16x4) x Q(4x16). D layout: lane = column i (lane%16), VGPR r =
// row j = r + 8*(lane/16). PV: O^T(dxi) += V-chunk(16x4) x P-pair(4x16), the
// P pairs come straight from S^T registers (contraction order {cr,cr+1,cr+8,cr+9}).
// V rows are replicated (row = Mv&3): D rows 4..15 are garbage but never read.
// Each lane-half keeps a partial row-sum l; merged once after the loop.
__global__ void __launch_bounds__(256) attn_kernel(const float* __restrict__ ws,
                                                   float* __restrict__ att) {
    const int lane = threadIdx.x & 31;
    const int wv   = threadIdx.x >> 5;
    const int tile = blockIdx.x * 8 + wv;          // 0..4095
    const int it   = tile & 255;
    const int bh   = tile >> 8;                    // b*4 + h
    const int Mv   = lane & 15;
    const int half = lane >> 4;
    const int i0   = it * 16;

    const float* qt = ws +              (size_t)bh * N_TOK * DHEAD;
    const float* kt = ws + REGION     + (size_t)bh * N_TOK * DHEAD;
    const float* vb = ws + 2 * REGION + (size_t)bh * DHEAD * N_TOK;

    // Q as B operand: lane needs q[i0+Mv][2*half .. 2*half+1] (contiguous pair)
    v2f bq = *(const v2f*)(qt + (size_t)(i0 + Mv) * DHEAD + 2 * half);

    v8f o = {};
    float m = -3.0e38f, lh = 0.f;   // lh: this half's partial row-sum

    const float* krow = kt + (size_t)Mv * DHEAD + 2 * half;
    const float* vrow = vb + (size_t)(Mv & 3) * N_TOK + 8 * half;

    // software pipeline: preload block 0 (32 j's)
    v2f ak0 = *(const v2f*)(krow);
    v2f ak1 = *(const v2f*)(krow + 16 * DHEAD);
    v4f va0 = *(const v4f*)(vrow);
    v4f va1 = *(const v4f*)(vrow + 4);
    v4f va2 = *(const v4f*)(vrow + 16);
    v4f va3 = *(const v4f*)(vrow + 20);

    for (int j0 = 0; j0 < N_TOK; j0 += 32) {
        v2f a0 = ak0, a1 = ak1;
        v4f w0 = va0, w1 = va1, w2 = va2, w3 = va3;
        // prefetch next block (stays inside the workspace even past the end)
        int jn = j0 + 32;
        ak0 = *(const v2f*)(krow + (size_t)jn * DHEAD);
        ak1 = *(const v2f*)(krow + (size_t)(jn + 16) * DHEAD);
        va0 = *(const v4f*)(vrow + jn);
        va1 = *(const v4f*)(vrow + jn + 4);
        va2 = *(const v4f*)(vrow + jn + 16);
        va3 = *(const v4f*)(vrow + jn + 20);

        v8f z = {};
        v8f sa = __builtin_amdgcn_wmma_f32_16x16x4_f32(false, a0, false, bq, (short)0, z, false, false);
        v8f sb = __builtin_amdgcn_wmma_f32_16x16x4_f32(false, a1, false, bq, (short)0, z, false, false);

        // combined tile max over 16 in-lane values + one cross-half swap (VALU)
        float t0 = fmaxf(fmaxf(fmaxf(sa[0], sa[1]), fmaxf(sa[2], sa[3])),
                         fmaxf(fmaxf(sa[4], sa[5]), fmaxf(sa[6], sa[7])));
        float t1 = fmaxf(fmaxf(fmaxf(sb[0], sb[1]), fmaxf(sb[2], sb[3])),
                         fmaxf(fmaxf(sb[4], sb[5]), fmaxf(sb[6], sb[7])));
        float tm = fmaxf(t0, t1);
        tm = fmaxf(tm, swap16(tm));
        float mn    = fmaxf(m, tm);
        float alpha = fexp2(m - mn);
        m = mn;

        float pa[8], pb_[8];
#pragma unroll
        for (int r = 0; r < 8; ++r) pa[r]  = fexp2(sa[r] - mn);
#pragma unroll
        for (int r = 0; r < 8; ++r) pb_[r] = fexp2(sb[r] - mn);
        float rs = (((pa[0] + pa[1]) + (pa[2] + pa[3])) + ((pa[4] + pa[5]) + (pa[6] + pa[7])))
                 + (((pb_[0] + pb_[1]) + (pb_[2] + pb_[3])) + ((pb_[4] + pb_[5]) + (pb_[6] + pb_[7])));
        lh = lh * alpha + rs;
#pragma unroll
        for (int r = 0; r < 8; ++r) o[r] *= alpha;

        // PV: 8 WMMAs, contraction order per chunk cr: {cr, cr+1, cr+8, cr+9}
        v2f av, pv;
        av.x = w0[0]; av.y = w0[1]; pv.x = pa[0]; pv.y = pa[1];
        o = __builtin_amdgcn_wmma_f32_16x16x4_f32(false, av, false, pv, (short)0, o, false, false);
        av.x = w0[2]; av.y = w0[3]; pv.x = pa[2]; pv.y = pa[3];
        o = __builtin_amdgcn_wmma_f32_16x16x4_f32(false, av, false, pv, (short)0, o, false, false);
        av.x = w1[0]; av.y = w1[1]; pv.x = pa[4]; pv.y = pa[5];
        o = __builtin_amdgcn_wmma_f32_16x16x4_f32(false, av, false, pv, (short)0, o, false, false);
        av.x = w1[2]; av.y = w1[3]; pv.x = pa[6]; pv.y = pa[7];
        o = __builtin_amdgcn_wmma_f32_16x16x4_f32(false, av, false, pv, (short)0, o, false, false);
        av.x = w2[0]; av.y = w2[1]; pv.x = pb_[0]; pv.y = pb_[1];
        o = __builtin_amdgcn_wmma_f32_16x16x4_f32(false, av, false, pv, (short)0, o, false, false);
        av.x = w2[2]; av.y = w2[3]; pv.x = pb_[2]; pv.y = pb_[3];
        o = __builtin_amdgcn_wmma_f32_16x16x4_f32(false, av, false, pv, (short)0, o, false, false);
        av.x = w3[0]; av.y = w3[1]; pv.x = pb_[4]; pv.y = pb_[5];
        o = __builtin_amdgcn_wmma_f32_16x16x4_f32(false, av, false, pv, (short)0, o, false, false);
        av.x = w3[2]; av.y = w3[3]; pv.x = pb_[6]; pv.y = pb_[7];
        o = __builtin_amdgcn_wmma_f32_16x16x4_f32(false, av, false, pv, (short)0, o, false, false);
    }

    float l = lh + swap16(lh);      // merge the two halves' partial sums once
    float inv = 1.0f / l;
    if (half == 0) {                // O^T rows d=0..3 live in VGPRs 0..3, lanes 0..15
        int b = bh >> 2, h = bh & 3;
        v4f st;
        st.x = o[0] * inv; st.y = o[1] * inv; st.z = o[2] * inv; st.w = o[3] * inv;
        // ATT layout [b][n][hd]: channels h*4..h*4+3 contiguous -> one b128 store
        *(v4f*)(att + ((size_t)b * N_TOK + i0 + Mv) * HIDDEN + h * DHEAD) = st;
    }
}

// ---------------- Kernel 3: y = w_out(64x16) @ att(B,n,16) + b_out ----------------
__global__ void __launch_bounds__(256) proj_kernel(const float* __restrict__ att,
                                                   const float* __restrict__ w_out,
                                                   const float* __restrict__ b_out,
                                                   float* __restrict__ y) {
    int tid = blockIdx.x * blockDim.x + threadIdx.x;  // = b*64*4096 + o*4096 + n
    int n = tid & (N_TOK - 1);
    int o = (tid >> 12) & 63;
    int b = tid >> 18;
    const float* ar = att + ((size_t)b * N_TOK + n) * HIDDEN;  // 16 contiguous floats
    const float* wr = w_out + o * HIDDEN;
    v4f a0 = *(const v4f*)(ar + 0);
    v4f a1 = *(const v4f*)(ar + 4);
    v4f a2 = *(const v4f*)(ar + 8);
    v4f a3 = *(const v4f*)(ar + 12);
    float acc = b_out[o];
#pragma unroll
    for (int c = 0; c < 4; ++c) acc = fmaf(wr[c],      a0[c], acc);
#pragma unroll
    for (int c = 0; c < 4; ++c) acc = fmaf(wr[4 + c],  a1[c], acc);
#pragma unroll
    for (int c = 0; c < 4; ++c) acc = fmaf(wr[8 + c],  a2[c], acc);
#pragma unroll
    for (int c = 0; c < 4; ++c) acc = fmaf(wr[12 + c], a3[c], acc);
    y[tid] = acc;
}

extern "C" void kernel_launch(void* const* d_in, const int* in_sizes, int n_in,
                              void* d_out, int out_size, void* d_ws, size_t ws_size,
                              hipStream_t stream) {
    const float* x     = (const float*)d_in[0];
    const float* w_qkv = (const float*)d_in[1];
    const float* w_out = (const float*)d_in[2];
    const float* b_out = (const float*)d_in[3];

    float* ws  = (float*)d_ws;                 // QT | KT | V  (3 * REGION floats)
    float* att = ws + 3 * (size_t)REGION;      // ATT (REGION floats)

    qkv_kernel<<<(BATCH * 48 * N_TOK) / 256, 256, 0, stream>>>(x, w_qkv, ws);
    attn_kernel<<<512, 256, 0, stream>>>(ws, att);
    proj_kernel<<<(BATCH * 64 * N_TOK) / 256, 256, 0, stream>>>(att, w_out, b_out, (float*)d_out);
}